// EnhancedQuantumLayer_33182917329657
// MI455X (gfx1250) — compile-verified
//
#include <hip/hip_runtime.h>
#include <math.h>

typedef __attribute__((ext_vector_type(2))) float v2f;
typedef __attribute__((ext_vector_type(8))) float v8f;

#define N_QUBITS 10
#define N_LAYERS 4
#define BATCH    512
#define BPW      16          // batch elements per workgroup (== WMMA M tile)
#define NSTATE   1024
#define PI_F     3.14159265358979323846f

struct c32 { float x, y; };

__device__ __forceinline__ c32 cmul(c32 a, c32 b) {
    return { a.x * b.x - a.y * b.y, a.x * b.y + a.y * b.x };
}
__device__ __forceinline__ c32 cadd(c32 a, c32 b) { return { a.x + b.x, a.y + b.y }; }

__global__ __launch_bounds__(512)
void qsim_kernel(const float* __restrict__ inputs,
                 const float* __restrict__ theta,
                 float* __restrict__ out) {
    extern __shared__ c32 st[];                       // [BPW][NSTATE] = 128 KB dynamic LDS
    __shared__ c32 vv[BPW][N_QUBITS][2];              // fused-encoding column vectors
    __shared__ c32 gg[N_LAYERS][N_QUBITS][4];         // fused layer 2x2 gates
    __shared__ float accp[256];                       // 16x16 WMMA partial accumulator

    const int t    = threadIdx.x;
    const int wg   = blockIdx.x;
    const int lane = t & 31;
    const int wave = t >> 5;                          // wave == batch-in-tile for gate phase

    // ---- Phase 0a: fused encoding vector per (batch, qubit): v = Pi_f (RX_f * RZ_f) |0>
    if (t < BPW * N_QUBITS) {
        int b = t / N_QUBITS, q = t % N_QUBITS;
        float xi = tanhf(inputs[(wg * BPW + b) * N_QUBITS + q]);
        c32 u0 = { 1.f, 0.f }, u1 = { 0.f, 0.f };
        #pragma unroll
        for (int fi = 0; fi < 5; ++fi) {
            float f = (float)(1 << fi);
            // RZ(f*xi): u0 *= e^{-i h}, u1 *= e^{+i h}, h = f*xi/2
            float h = 0.5f * f * xi;
            float cz = cosf(h), sz = sinf(h);
            u0 = cmul(u0, { cz, -sz });
            u1 = cmul(u1, { cz,  sz });
            // RX(f*xi*0.5): half-angle = f*xi/4
            float hr = 0.25f * f * xi;
            float cr = cosf(hr), sr = sinf(hr);
            c32 n0 = { cr * u0.x + sr * u1.y,  cr * u0.y - sr * u1.x };
            c32 n1 = { sr * u0.y + cr * u1.x, -sr * u0.x + cr * u1.y };
            u0 = n0; u1 = n1;
        }
        vv[b][q][0] = u0; vv[b][q][1] = u1;
    }
    // ---- Phase 0b: fused layer gates G = RX(a0/2)*RZ(a2)*RY(a1)*RZ(a0)
    if (t >= 256 && t < 256 + N_LAYERS * N_QUBITS) {
        int i = t - 256;
        int l = i / N_QUBITS, q = i % N_QUBITS;
        int base = (l * N_QUBITS + q) * 3;
        float a0 = tanhf(theta[base + 0]) * PI_F;
        float a1 = tanhf(theta[base + 1]) * PI_F;
        float a2 = tanhf(theta[base + 2]) * PI_F;
        // M = RZ(a0)
        float c0 = cosf(0.5f * a0), s0 = sinf(0.5f * a0);
        c32 m00 = { c0, -s0 }, m01 = { 0.f, 0.f }, m10 = { 0.f, 0.f }, m11 = { c0, s0 };
        // M = RY(a1) * M   (real rotation)
        float cy = cosf(0.5f * a1), sy = sinf(0.5f * a1);
        c32 r00 = { cy * m00.x - sy * m10.x, cy * m00.y - sy * m10.y };
        c32 r01 = { cy * m01.x - sy * m11.x, cy * m01.y - sy * m11.y };
        c32 r10 = { sy * m00.x + cy * m10.x, sy * m00.y + cy * m10.y };
        c32 r11 = { sy * m01.x + cy * m11.x, sy * m01.y + cy * m11.y };
        m00 = r00; m01 = r01; m10 = r10; m11 = r11;
        // M = RZ(a2) * M   (row phases)
        float c2 = cosf(0.5f * a2), s2 = sinf(0.5f * a2);
        m00 = cmul({ c2, -s2 }, m00); m01 = cmul({ c2, -s2 }, m01);
        m10 = cmul({ c2,  s2 }, m10); m11 = cmul({ c2,  s2 }, m11);
        // M = RX(a0*0.5) * M  (half-angle a0/4);  (-i*s)*z = (s*z.y, -s*z.x)
        float cx = cosf(0.25f * a0), sx = sinf(0.25f * a0);
        c32 t00 = { cx * m00.x + sx * m10.y,  cx * m00.y - sx * m10.x };
        c32 t01 = { cx * m01.x + sx * m11.y,  cx * m01.y - sx * m11.x };
        c32 t10 = { sx * m00.y + cx * m10.x, -sx * m00.x + cx * m10.y };
        c32 t11 = { sx * m01.y + cx * m11.x, -sx * m01.x + cx * m11.y };
        gg[l][q][0] = t00; gg[l][q][1] = t01; gg[l][q][2] = t10; gg[l][q][3] = t11;
    }
    if (t < 256) accp[t] = 0.f;
    __syncthreads();

    // ---- Phase 1: tensor-product initialization of the state (qubit 0 = MSB of index)
    {
        const int b = wave;
        for (int j = 0; j < 32; ++j) {
            int s = lane + 32 * j;
            c32 amp = vv[b][0][(s >> 9) & 1];
            #pragma unroll
            for (int q = 1; q < N_QUBITS; ++q)
                amp = cmul(amp, vv[b][q][(s >> (9 - q)) & 1]);
            st[b * NSTATE + s] = amp;
        }
    }
    __syncthreads();

    // ---- Phase 2: variational layers, one wave per batch element, state in LDS
    {
        const int b = wave;
        for (int l = 0; l < N_LAYERS; ++l) {
            for (int q = 0; q < N_QUBITS; ++q) {       // fused 1q gate on qubit q
                c32 g00 = gg[l][q][0], g01 = gg[l][q][1];
                c32 g10 = gg[l][q][2], g11 = gg[l][q][3];
                int p = 9 - q;
                int lowmask = (1 << p) - 1;
                for (int j = 0; j < 16; ++j) {
                    int k  = lane + 32 * j;
                    int i0 = ((k >> p) << (p + 1)) | (k & lowmask);
                    int i1 = i0 | (1 << p);
                    c32 a0 = st[b * NSTATE + i0];
                    c32 a1 = st[b * NSTATE + i1];
                    c32 n0 = cadd(cmul(g00, a0), cmul(g01, a1));
                    c32 n1 = cadd(cmul(g10, a0), cmul(g11, a1));
                    st[b * NSTATE + i0] = n0;
                    st[b * NSTATE + i1] = n1;
                }
                __syncthreads();
            }
            // entanglers: signed swap on (c=1) subspace; out|10> = -in|11>, out|11> = in|10>
            for (int parity = 0; parity < 2; ++parity) {
                for (int q = parity; q < N_QUBITS - 1; q += 2) {
                    int pt = 8 - q;                    // target bit; control bit = pt+1
                    int lowmask = (1 << pt) - 1;
                    for (int j = 0; j < 8; ++j) {
                        int k   = lane + 32 * j;
                        int ib  = ((k >> pt) << (pt + 2)) | (k & lowmask);
                        int s10 = ib | (2 << pt);
                        int s11 = s10 | (1 << pt);
                        c32 a  = st[b * NSTATE + s10];
                        c32 bb = st[b * NSTATE + s11];
                        st[b * NSTATE + s10] = { -bb.x, -bb.y };
                        st[b * NSTATE + s11] = a;
                    }
                    __syncthreads();
                }
            }
        }
    }

    // ---- Phase 3: readout GEMM  out[m][n] = sum_k p[m][k] * Z[k][n]  via f32 WMMA 16x16x4
    // Each wave owns a 64-wide K slice (16 WMMA steps); partial D tiles merged via ds_add_f32.
    {
        const int mrow = lane & 15;                    // A-operand M row (batch in tile)
        const int nq   = lane & 15;                    // B/D N column (qubit)
        const int hi   = (lane >> 4) & 1;              // low lanes: K=0,1  high lanes: K=2,3
        v8f acc = { 0.f, 0.f, 0.f, 0.f, 0.f, 0.f, 0.f, 0.f };
        for (int c = 0; c < 16; ++c) {
            int k0 = wave * 64 + c * 4 + hi * 2;
            c32 s0 = st[mrow * NSTATE + k0];
            c32 s1 = st[mrow * NSTATE + k0 + 1];
            v2f A;                                      // probabilities
            A.x = s0.x * s0.x + s0.y * s0.y;
            A.y = s1.x * s1.x + s1.y * s1.y;
            v2f B;                                      // Z-sign matrix (+1/-1), 0-padded cols
            B.x = (nq < N_QUBITS) ? (float)(1 - 2 * ((k0 >> (9 - nq)) & 1)) : 0.f;
            B.y = (nq < N_QUBITS) ? (float)(1 - 2 * (((k0 + 1) >> (9 - nq)) & 1)) : 0.f;
            acc = __builtin_amdgcn_wmma_f32_16x16x4_f32(
                false, A, false, B, (short)0, acc, false, false);
        }
        #pragma unroll
        for (int r = 0; r < 8; ++r) {
            int m = r + hi * 8;                         // D layout: VGPR r -> M=r (+8 hi half)
            atomicAdd(&accp[m * 16 + nq], acc[r]);
        }
    }
    __syncthreads();

    if (t < BPW * N_QUBITS) {
        int m = t / N_QUBITS, n = t % N_QUBITS;
        out[(wg * BPW + m) * N_QUBITS + n] = accp[m * 16 + n];
    }
}

extern "C" void kernel_launch(void* const* d_in, const int* in_sizes, int n_in,
                              void* d_out, int out_size, void* d_ws, size_t ws_size,
                              hipStream_t stream) {
    (void)in_sizes; (void)n_in; (void)out_size; (void)d_ws; (void)ws_size;
    const float* inputs = (const float*)d_in[0];   // (512, 10) f32
    const float* theta  = (const float*)d_in[1];   // (4, 10, 3) f32
    float* out = (float*)d_out;                    // (512, 10) f32
    size_t shmem = (size_t)BPW * NSTATE * sizeof(c32);   // 128 KB of the 320 KB WGP LDS
    hipLaunchKernelGGL(qsim_kernel, dim3(BATCH / BPW), dim3(512), shmem, stream,
                       inputs, theta, out);
}